// CGNonLinearity3D_7310034337923
// MI455X (gfx1250) — compile-verified
//
#include <hip/hip_runtime.h>
#include <hip/hip_bf16.h>

typedef float v2f __attribute__((ext_vector_type(2)));
typedef float v8f __attribute__((ext_vector_type(8)));

// ---------------------------------------------------------------------------
// Compile-time Clebsch-Gordan tables (exact transcription of the reference's
// Racah formula, evaluated in double at C++ compile time).
// ---------------------------------------------------------------------------
namespace cg {

constexpr double fact(int n) {
  double r = 1.0;
  for (int i = 2; i <= n; ++i) r *= (double)i;
  return r;
}

constexpr double csqrt(double x) {
  if (x <= 0.0) return 0.0;
  double g = x > 1.0 ? x : 1.0;
  for (int i = 0; i < 64; ++i) g = 0.5 * (g + x / g);
  return g;
}

constexpr double coeff(int j1, int m1, int j2, int m2, int j, int m) {
  if (m1 + m2 != m) return 0.0;
  int dj = j1 - j2; if (dj < 0) dj = -dj;
  if (j < dj || j > j1 + j2) return 0.0;
  double pref = csqrt((double)(2 * j + 1) * fact(j + j1 - j2) * fact(j - j1 + j2) *
                      fact(j1 + j2 - j) / fact(j1 + j2 + j + 1));
  pref *= csqrt(fact(j + m) * fact(j - m) * fact(j1 - m1) * fact(j1 + m1) *
                fact(j2 - m2) * fact(j2 + m2));
  int kmin = 0;
  if (j2 - j - m1 > kmin) kmin = j2 - j - m1;
  if (j1 - j + m2 > kmin) kmin = j1 - j + m2;
  int kmax = j1 + j2 - j;
  if (j1 - m1 < kmax) kmax = j1 - m1;
  if (j2 + m2 < kmax) kmax = j2 + m2;
  double s = 0.0;
  for (int k = kmin; k <= kmax; ++k) {
    double d = fact(k) * fact(j1 + j2 - j - k) * fact(j1 - m1 - k) *
               fact(j2 + m2 - k) * fact(j - j2 + m1 + k) * fact(j - j1 - m2 + k);
    s += ((k & 1) ? -1.0 : 1.0) / d;
  }
  return pref * s;
}

// (l1,l2) pairs per output l, l1-outer / l2-inner (matches reference _PAIRS).
constexpr int NPAIRS[3] = {3, 6, 6};
constexpr int PL1[3][6] = {{0, 1, 2, 0, 0, 0}, {0, 1, 1, 1, 2, 2}, {0, 1, 1, 2, 2, 2}};
constexpr int PL2[3][6] = {{0, 1, 2, 0, 0, 0}, {1, 0, 1, 2, 1, 2}, {2, 1, 2, 0, 1, 2}};

struct Term { int i; int j; float c; };
struct Tables {
  Term terms[160];
  int off[3][6][5];
  int cnt[3][6][5];
};

constexpr Tables make() {
  Tables T{};
  int t = 0;
  for (int l = 0; l < 3; ++l)
    for (int p = 0; p < NPAIRS[l]; ++p) {
      const int l1 = PL1[l][p], l2 = PL2[l][p];
      for (int mi = 0; mi < 2 * l + 1; ++mi) {
        T.off[l][p][mi] = t;
        const int m = mi - l;
        for (int i = 0; i < 2 * l1 + 1; ++i)
          for (int jj = 0; jj < 2 * l2 + 1; ++jj) {
            const int m1 = i - l1, m2 = jj - l2;
            if (m1 + m2 != m) continue;
            const double c = coeff(l1, m1, l2, m2, l, m);
            if (c != 0.0) { T.terms[t].i = i; T.terms[t].j = jj; T.terms[t].c = (float)c; ++t; }
          }
        T.cnt[l][p][mi] = t - T.off[l][p][mi];
      }
    }
  return T;
}

constexpr Tables TAB = make();

}  // namespace cg

// ---------------------------------------------------------------------------
// Fully unrolled CG accumulation: every coefficient is a literal immediate.
// ---------------------------------------------------------------------------
template <int OFF, int CNT>
__device__ __forceinline__ void accum_terms(const float (&ar)[5][2], const float (&br)[5][2],
                                            float& sre, float& sim) {
  if constexpr (CNT > 0) {
    constexpr cg::Term tm = cg::TAB.terms[OFF];
    const float xr = ar[tm.i][0], xi = ar[tm.i][1];
    const float yr = br[tm.j][0], yi = br[tm.j][1];
    sre = __builtin_fmaf(tm.c, xr * yr - xi * yi, sre);
    sim = __builtin_fmaf(tm.c, xr * yi + xi * yr, sim);
    accum_terms<OFF + 1, CNT - 1>(ar, br, sre, sim);
  }
}

// ---------------------------------------------------------------------------
// One (l1,l2) pair: build 16(b) x 256(k) complex cat-fragment in LDS, then
// contract the 256-wide K block against w with f32 WMMA (complex = 4 matmuls
// per n-subtile; each wave owns two n-subtiles so each A fragment feeds 8).
// ---------------------------------------------------------------------------
template <int L, int M, int P>
__device__ __forceinline__ void pair_step(const float* const (&xs)[3], const float* __restrict__ w,
                                          float (&lre)[16][257], float (&lim)[16][257],
                                          v8f& cre0, v8f& cim0, v8f& cre1, v8f& cim1,
                                          int b0, int n0, int lane) {
  constexpr int l1 = cg::PL1[L][P];
  constexpr int l2 = cg::PL2[L][P];
  constexpr int OFF = cg::TAB.off[L][P][M];
  constexpr int CNT = cg::TAB.cnt[L][P][M];

  const int k = threadIdx.x;  // 256 threads: one channel each
  const float* __restrict__ xa = xs[l1];
  const float* __restrict__ xb = xs[l2];

  __syncthreads();  // previous pair's LDS reads are done
  #pragma unroll 4
  for (int b = 0; b < 16; ++b) {
    float ar[5][2];
    float br[5][2];
    const float* pa = xa + ((size_t)(b0 + b) * (2 * l1 + 1) * 256 + k) * 2;
    #pragma unroll
    for (int i = 0; i < 2 * l1 + 1; ++i) { ar[i][0] = pa[0]; ar[i][1] = pa[1]; pa += 512; }
    const float* pb = xb + ((size_t)(b0 + b) * (2 * l2 + 1) * 256 + k) * 2;
    #pragma unroll
    for (int j = 0; j < 2 * l2 + 1; ++j) { br[j][0] = pb[0]; br[j][1] = pb[1]; pb += 512; }
    float sre = 0.f, sim = 0.f;
    accum_terms<OFF, CNT>(ar, br, sre, sim);
    lre[b][k] = sre;
    lim[b][k] = sim;
  }
  __syncthreads();  // cat fragment ready

  const int bl = lane & 15;          // A: row (b-local);  B/D: column (n)
  const int kh = (lane >> 4) << 1;   // K sub-offset per half-wave
  for (int kc = 0; kc < 64; ++kc) {
    const int kk = (kc << 2) + kh;
    v2f are = {lre[bl][kk], lre[bl][kk + 1]};
    v2f aim = {lim[bl][kk], lim[bl][kk + 1]};
    v2f nim = {-aim.x, -aim.y};
    // w element (K, n) complex at (K*256 + n)*2; K = P*256 + kk (+1)
    const float* wp = w + ((size_t)(P * 256 + kk) * 256 + (n0 + bl)) * 2;
    {  // n-subtile 0: columns [n0, n0+16)
      const float w0r = wp[0], w0i = wp[1];
      const float w1r = wp[512], w1i = wp[513];
      v2f bre = {w0r, w1r};
      v2f bim = {w0i, w1i};
      cre0 = __builtin_amdgcn_wmma_f32_16x16x4_f32(false, are, false, bre, (short)0, cre0, false, false);
      cre0 = __builtin_amdgcn_wmma_f32_16x16x4_f32(false, nim, false, bim, (short)0, cre0, false, false);
      cim0 = __builtin_amdgcn_wmma_f32_16x16x4_f32(false, are, false, bim, (short)0, cim0, false, false);
      cim0 = __builtin_amdgcn_wmma_f32_16x16x4_f32(false, aim, false, bre, (short)0, cim0, false, false);
    }
    {  // n-subtile 1: columns [n0+128, n0+144)
      const float w0r = wp[256], w0i = wp[257];
      const float w1r = wp[768], w1i = wp[769];
      v2f bre = {w0r, w1r};
      v2f bim = {w0i, w1i};
      cre1 = __builtin_amdgcn_wmma_f32_16x16x4_f32(false, are, false, bre, (short)0, cre1, false, false);
      cre1 = __builtin_amdgcn_wmma_f32_16x16x4_f32(false, nim, false, bim, (short)0, cre1, false, false);
      cim1 = __builtin_amdgcn_wmma_f32_16x16x4_f32(false, are, false, bim, (short)0, cim1, false, false);
      cim1 = __builtin_amdgcn_wmma_f32_16x16x4_f32(false, aim, false, bre, (short)0, cim1, false, false);
    }
  }
}

template <int L, int M, int P>
__device__ __forceinline__ void pairs_loop(const float* const (&xs)[3], const float* __restrict__ w,
                                           float (&lre)[16][257], float (&lim)[16][257],
                                           v8f& cre0, v8f& cim0, v8f& cre1, v8f& cim1,
                                           int b0, int n0, int lane) {
  if constexpr (P < cg::NPAIRS[L]) {
    pair_step<L, M, P>(xs, w, lre, lim, cre0, cim0, cre1, cim1, b0, n0, lane);
    pairs_loop<L, M, P + 1>(xs, w, lre, lim, cre0, cim0, cre1, cim1, b0, n0, lane);
  }
}

// ---------------------------------------------------------------------------
// Kernel: out[b, M, n] for one (L, M); tile = 16 b-rows x 256 n-cols.
// 256 threads = 8 waves; wave w owns columns [16w,16w+16) and [16w+128,16w+144).
// ---------------------------------------------------------------------------
template <int L, int M>
__global__ __launch_bounds__(256)
void cg_gemm(const float* __restrict__ x0, const float* __restrict__ x1,
             const float* __restrict__ x2, const float* __restrict__ w,
             float* __restrict__ out) {
  __shared__ float lre[16][257];  // +1 pad: kills 64-bank conflicts on column reads
  __shared__ float lim[16][257];

  const int lane = threadIdx.x & 31;
  const int wv = threadIdx.x >> 5;
  const int b0 = blockIdx.x * 16;
  const int n0 = wv * 16;

  const float* const xs[3] = {x0, x1, x2};

  v8f cre0 = {};
  v8f cim0 = {};
  v8f cre1 = {};
  v8f cim1 = {};

  pairs_loop<L, M, 0>(xs, w, lre, lim, cre0, cim0, cre1, cim1, b0, n0, lane);

  // D layout: VGPR r <-> tile row r + 8*(lane>=16), col = lane&15
  const int bl = lane & 15;
  const int rb = (lane >> 4) * 8;
  #pragma unroll
  for (int r = 0; r < 8; ++r) {
    const int b = b0 + rb + r;
    float* po = out + ((size_t)b * (2 * L + 1) + M) * 512 + (n0 + bl) * 2;
    po[0] = cre0[r];
    po[1] = cim0[r];
    po[256] = cre1[r];
    po[257] = cim1[r];
  }
}

// ---------------------------------------------------------------------------
// Launch: inputs in dict order x0, w0, x1, w1, x2, w2.
// ---------------------------------------------------------------------------
extern "C" void kernel_launch(void* const* d_in, const int* in_sizes, int n_in,
                              void* d_out, int out_size, void* d_ws, size_t ws_size,
                              hipStream_t stream) {
  (void)n_in; (void)out_size; (void)d_ws; (void)ws_size;
  const float* x0 = (const float*)d_in[0];
  const float* w0 = (const float*)d_in[1];
  const float* x1 = (const float*)d_in[2];
  const float* w1 = (const float*)d_in[3];
  const float* x2 = (const float*)d_in[4];
  const float* w2 = (const float*)d_in[5];
  float* out = (float*)d_out;

  const int B = in_sizes[0] / 512;  // x0 = B*1*256*2 floats
  dim3 grid(B / 16, 1);
  dim3 block(256);

  float* out0 = out;
  float* out1 = out0 + (size_t)B * 1 * 256 * 2;
  float* out2 = out1 + (size_t)B * 3 * 256 * 2;

  cg_gemm<0, 0><<<grid, block, 0, stream>>>(x0, x1, x2, w0, out0);

  cg_gemm<1, 0><<<grid, block, 0, stream>>>(x0, x1, x2, w1, out1);
  cg_gemm<1, 1><<<grid, block, 0, stream>>>(x0, x1, x2, w1, out1);
  cg_gemm<1, 2><<<grid, block, 0, stream>>>(x0, x1, x2, w1, out1);

  cg_gemm<2, 0><<<grid, block, 0, stream>>>(x0, x1, x2, w2, out2);
  cg_gemm<2, 1><<<grid, block, 0, stream>>>(x0, x1, x2, w2, out2);
  cg_gemm<2, 2><<<grid, block, 0, stream>>>(x0, x1, x2, w2, out2);
  cg_gemm<2, 3><<<grid, block, 0, stream>>>(x0, x1, x2, w2, out2);
  cg_gemm<2, 4><<<grid, block, 0, stream>>>(x0, x1, x2, w2, out2);
}